// EmbeddingLayer_56521769615468
// MI455X (gfx1250) — compile-verified
//
#include <hip/hip_runtime.h>
#include <cstdint>
#include <cstddef>

#define N_SPARSE 16
#define N_SEQ 2
#define N_DENSE 13
#define VOCAB 100000
#define EMBED 64
#define SEQ_LEN 50
#define BATCH 4096
#define OUT_STRIDE ((N_SPARSE + N_SEQ) * EMBED + N_DENSE) /* 1165 */

typedef float v4f __attribute__((ext_vector_type(4)));

// One block (64 threads = 2 wave32) per batch row.
//  1) load all gather indices into registers (independent, pipelined loads)
//  2) fire 25 GLOBAL_LOAD_ASYNC_TO_LDS_B128 in GVS mode (saddr = table base,
//     u32 voffset; 100 seq rows -> 25.6KB LDS)
//  3) overlap: sparse gathers (NT loads) + dense tail -> out (NT stores)
//  4) s_wait_asynccnt 0, barrier, pool from LDS, NT-store results
__global__ __launch_bounds__(64) void embedding_layer_kernel(
    const float* __restrict__ sparse_tables,  // [16, VOCAB, 64]
    const float* __restrict__ seq_tables,     // [ 2, VOCAB, 64]
    const float* __restrict__ dense_vals,     // [B, 13]
    const int*   __restrict__ sparse_idx,     // [16, B]
    const int*   __restrict__ seq_idx,        // [2, B, 50]
    float* __restrict__ out)                  // [B, 1165]
{
    const int b = blockIdx.x;
    const int t = threadIdx.x;

    __shared__ __align__(16) float smem[(2 * SEQ_LEN) * EMBED]; // 100 rows x 256B

    // Wave-relative LDS byte offset of smem (low 32 bits of generic pointer).
    const unsigned lds_base = (unsigned)(uintptr_t)(&smem[0]);

    // Per-thread staging geometry: unit u = i*64 + t covers 16B.
    // chunk c = t & 15 is FIXED per thread; row r = (t>>4) + 4*i.
    const int c  = t & 15;        // 16B chunk within a 256B row
    const int r0 = t >> 4;        // starting row 0..3

    // ---- 1) preload every index BEFORE any asm (keeps loads pipelined) ----
    int seq_i[25];
#pragma unroll
    for (int i = 0; i < 25; ++i) {
        const int r    = r0 + 4 * i;                 // 0..99
        const int feat = (r >= SEQ_LEN) ? 1 : 0;
        const int s    = r - feat * SEQ_LEN;
        seq_i[i] = seq_idx[((size_t)feat * BATCH + (size_t)b) * SEQ_LEN + s];
    }

    int sp_i[4];
#pragma unroll
    for (int i = 0; i < 4; ++i) {
        const int f = (i * 64 + t) >> 4;             // feature 0..15
        sp_i[i] = sparse_idx[(size_t)f * BATCH + (size_t)b];
    }

    float dv = 0.0f;
    if (t < N_DENSE)
        dv = __builtin_nontemporal_load(dense_vals + (size_t)b * N_DENSE + t);

    // ---- 2) fire-and-forget async DMA of the 100 seq rows into LDS -------
    // GVS mode: mem = SGPR64(seq_tables) + u32 voffset. Both seq tables span
    // only 51.2MB, so a 32-bit offset covers everything (single 24-bit mad
    // per unit instead of 64-bit address chains).
    // 16 consecutive lanes cover one full 256B row (coalesced 512B/instr).
#pragma unroll
    for (int i = 0; i < 25; ++i) {
        const int r    = r0 + 4 * i;
        const int feat = (r >= SEQ_LEN) ? 1 : 0;
        const unsigned goff =
            ((unsigned)feat * VOCAB + (unsigned)seq_i[i]) * (EMBED * 4) +
            (unsigned)(c * 16);
        const unsigned lds_off = lds_base + (unsigned)(r * 256 + c * 16);
        asm volatile("global_load_async_to_lds_b128 %0, %1, %2"
                     :
                     : "v"(lds_off), "v"(goff), "s"(seq_tables));
    }

    const size_t obase = (size_t)b * OUT_STRIDE;

    // ---- 3) sparse gathers overlapped with DMA ----------------------------
    // Sparse tables are 410MB with ~0.04x reuse: non-temporal so the 51MB
    // seq tables (~4x reuse) stay resident in the 192MB L2.
#pragma unroll
    for (int i = 0; i < 4; ++i) {
        const int f = (i * 64 + t) >> 4;             // feature 0..15
        const v4f v = __builtin_nontemporal_load(reinterpret_cast<const v4f*>(
            sparse_tables + ((size_t)f * VOCAB + (size_t)sp_i[i]) * EMBED +
            (size_t)(c * 4)));
        float* o = out + obase + (size_t)(f * EMBED + c * 4);  // 4B-aligned only
        __builtin_nontemporal_store(v.x, o + 0);
        __builtin_nontemporal_store(v.y, o + 1);
        __builtin_nontemporal_store(v.z, o + 2);
        __builtin_nontemporal_store(v.w, o + 3);
    }

    if (t < N_DENSE)
        __builtin_nontemporal_store(
            dv, out + obase + (size_t)((N_SPARSE + N_SEQ) * EMBED) + t);

    // ---- 4) wait for this wave's DMA, then cross-wave barrier -------------
    asm volatile("s_wait_asynccnt 0x0" ::: "memory");
    __syncthreads();

    // thread t owns embedding dim d = t; stride-1 across lanes => no conflicts
    float sum0 = 0.0f;
#pragma unroll
    for (int s = 0; s < SEQ_LEN; ++s)
        sum0 += smem[s * EMBED + t];

    float sum1 = 0.0f, cnt = 0.0f;
#pragma unroll
    for (int s = 0; s < SEQ_LEN; ++s) {
        const float v = smem[(SEQ_LEN + s) * EMBED + t];
        sum1 += v;
        cnt  += (v != 0.0f) ? 1.0f : 0.0f;
    }

    __builtin_nontemporal_store(sum0, out + obase + (size_t)(N_SPARSE * EMBED) + t);
    __builtin_nontemporal_store(sum1 / (cnt + 1e-16f),
                                out + obase + (size_t)((N_SPARSE + 1) * EMBED) + t);
}

extern "C" void kernel_launch(void* const* d_in, const int* in_sizes, int n_in,
                              void* d_out, int out_size, void* d_ws, size_t ws_size,
                              hipStream_t stream) {
    const float* sparse_tables = (const float*)d_in[0];
    const float* seq_tables    = (const float*)d_in[1];
    const float* dense_vals    = (const float*)d_in[2];
    const int*   sparse_idx    = (const int*)d_in[3];
    const int*   seq_idx       = (const int*)d_in[4];
    float*       out           = (float*)d_out;

    embedding_layer_kernel<<<BATCH, 64, 0, stream>>>(
        sparse_tables, seq_tables, dense_vals, sparse_idx, seq_idx, out);
}